// TripletLoss_Cloth_Sen_37726992728636
// MI455X (gfx1250) — compile-verified
//
#include <hip/hip_runtime.h>

typedef float v2f __attribute__((ext_vector_type(2)));
typedef float v8f __attribute__((ext_vector_type(8)));

#define NPTS 8192
#define DIM  128
#define NUMC 64
#define MARGIN 0.3f
#define NTILE (NPTS / 16)            // 512 column tiles
#define WAVES 4                      // waves per block, one 16-row A panel each
#define ROWS_PER_BLOCK (16 * WAVES)  // 64
#define COLSPLIT 4                   // column-chunk blocks per row group
#define TILES_PER_BLOCK (NTILE / COLSPLIT)  // 128
#define BSTRIDE 132                  // padded LDS row stride (floats) -> bank-conflict-free b64 reads

// ---------------------------------------------------------------------------
// Kernel 1: per-row squared norms + init of the global max/min bit arrays
// ---------------------------------------------------------------------------
__global__ __launch_bounds__(256) void prep_kernel(const float* __restrict__ x,
                                                   float* __restrict__ sq,
                                                   unsigned* __restrict__ apb,
                                                   unsigned* __restrict__ anb) {
    const int row = blockIdx.x * 256 + threadIdx.x;
    const float4* p = (const float4*)(x + (size_t)row * DIM);
    float s = 0.0f;
#pragma unroll
    for (int i = 0; i < DIM / 4; ++i) {
        float4 v = p[i];
        s += v.x * v.x + v.y * v.y + v.z * v.z + v.w * v.w;
    }
    sq[row]  = s;
    apb[row] = 0u;           // max accumulator (distances >= 0)
    anb[row] = 0x7F800000u;  // min accumulator (+inf bits)
}

// ---------------------------------------------------------------------------
// Kernel 2: blocked fused Gram WMMA + masked row max/min.
// Block = 64 rows x 128 column tiles. B panel staged in LDS, shared by 4 waves.
// ---------------------------------------------------------------------------
__global__ __launch_bounds__(32 * WAVES) void tile_kernel(
    const float* __restrict__ x, const int* __restrict__ tgt,
    const float* __restrict__ sq,
    unsigned* __restrict__ apb, unsigned* __restrict__ anb) {

    __shared__ float lb[16 * BSTRIDE];  // B panel, padded rows
    __shared__ float lsq[16];
    __shared__ int   ltj[16];

    const int tid  = threadIdx.x;
    const int lane = tid & 31;
    const int wave = tid >> 5;
    const int lmod = lane & 15;  // M for A, N for B (ISA 16x4 fp32 layout)
    const int lhi  = lane >> 4;  // K-pair select
    const int koff = lhi * 2;

    const int rowbase = blockIdx.y * ROWS_PER_BLOCK + wave * 16;

    // --- preload this wave's A panel (16 rows x 128 K) into registers
    v2f af[32];
    const float* arow = x + (size_t)(rowbase + lmod) * DIM + koff;
#pragma unroll
    for (int kk = 0; kk < 32; ++kk)
        af[kk] = *(const v2f*)(arow + 4 * kk);

    int   tiv[8];
    float sqi[8];
#pragma unroll
    for (int v = 0; v < 8; ++v) {
        const int r = rowbase + v + 8 * lhi;
        tiv[v] = tgt[r];
        sqi[v] = sq[r];
    }

    const float INF = __uint_as_float(0x7F800000u);
    float rmax[8], rmin[8];
#pragma unroll
    for (int v = 0; v < 8; ++v) { rmax[v] = -INF; rmin[v] = INF; }

    // staging map: thread -> (row r_, 16-float segment seg) of the B panel
    const int r_  = tid >> 3;
    const int seg = (tid & 7) * 16;

    const int jt0 = blockIdx.x * TILES_PER_BLOCK;
    for (int jt = jt0; jt < jt0 + TILES_PER_BLOCK; ++jt) {
        const int colbase = jt * 16;
        __syncthreads();
        {
            const float* src = x + (size_t)(colbase + r_) * DIM + seg;
            float* dst = lb + r_ * BSTRIDE + seg;
#pragma unroll
            for (int u = 0; u < 4; ++u)
                *(float4*)(dst + 4 * u) = *(const float4*)(src + 4 * u);
            if (tid < 16) {
                ltj[tid] = tgt[colbase + tid];
                lsq[tid] = sq[colbase + tid];
            }
            if (jt + 1 < jt0 + TILES_PER_BLOCK)  // block-uniform condition
                __builtin_prefetch(x + (size_t)(colbase + 16 + r_) * DIM + seg, 0, 0);
        }
        __syncthreads();

        // --- 32-step fp32 WMMA chain against the shared LDS panel
        const float* bp = lb + lmod * BSTRIDE + koff;
        v8f c = {0.f, 0.f, 0.f, 0.f, 0.f, 0.f, 0.f, 0.f};
#pragma unroll
        for (int kk = 0; kk < 32; ++kk) {
            v2f bf = *(const v2f*)(bp + 4 * kk);
            c = __builtin_amdgcn_wmma_f32_16x16x4_f32(
                    false, af[kk], false, bf, (short)0, c, false, false);
        }

        const int   tj  = ltj[lmod];
        const float sqj = lsq[lmod];
#pragma unroll
        for (int v = 0; v < 8; ++v) {
            float d2 = sqi[v] + sqj - 2.0f * c[v];
            float d  = __builtin_sqrtf(fmaxf(d2, 1e-12f));
            const int ti = tiv[v];
            const bool pos = (tj == ti);
            const bool pn  = (tj == ti + NUMC) || (tj == ti - NUMC);
            rmax[v] = pos ? fmaxf(rmax[v], d) : rmax[v];
            rmin[v] = pn  ? fminf(rmin[v], d) : rmin[v];
        }
    }

    // --- reduce across the 16 lanes sharing each row (xor 1,2,4,8 stays in half)
#pragma unroll
    for (int m = 1; m <= 8; m <<= 1) {
#pragma unroll
        for (int v = 0; v < 8; ++v) {
            rmax[v] = fmaxf(rmax[v], __shfl_xor(rmax[v], m, 32));
            rmin[v] = fminf(rmin[v], __shfl_xor(rmin[v], m, 32));
        }
    }

    // --- merge into global per-row accumulators (order-independent bit atomics)
    if (lmod == 0) {
#pragma unroll
        for (int v = 0; v < 8; ++v) {
            const int r = rowbase + v + 8 * lhi;
            atomicMax(&apb[r], __float_as_uint(fmaxf(rmax[v], 0.0f)));
            atomicMin(&anb[r], __float_as_uint(rmin[v]));
        }
    }
}

// ---------------------------------------------------------------------------
// Kernel 3: deterministic fixed-order loss reduction -> mean
// ---------------------------------------------------------------------------
__global__ __launch_bounds__(256) void final_kernel(const unsigned* __restrict__ apb,
                                                    const unsigned* __restrict__ anb,
                                                    float* __restrict__ out) {
    __shared__ float red[256];
    const int tid = threadIdx.x;
    float s = 0.0f;
#pragma unroll 4
    for (int i = 0; i < NPTS / 256; ++i) {
        const int r = tid * (NPTS / 256) + i;
        const float ap = __uint_as_float(apb[r]);
        const float an = __uint_as_float(anb[r]);
        s += fmaxf(ap - an + MARGIN, 0.0f);  // an=+inf -> contributes 0
    }
    red[tid] = s;
    __syncthreads();
    for (int off = 128; off > 0; off >>= 1) {
        if (tid < off) red[tid] += red[tid + off];
        __syncthreads();
    }
    if (tid == 0) out[0] = red[0] / (float)NPTS;
}

// ---------------------------------------------------------------------------
extern "C" void kernel_launch(void* const* d_in, const int* in_sizes, int n_in,
                              void* d_out, int out_size, void* d_ws, size_t ws_size,
                              hipStream_t stream) {
    const float* x   = (const float*)d_in[0];
    const int*   tgt = (const int*)d_in[1];
    float* out = (float*)d_out;

    float*    sq  = (float*)d_ws;              // 8192 floats
    unsigned* apb = (unsigned*)(sq + NPTS);    // 8192 uints
    unsigned* anb = apb + NPTS;                // 8192 uints

    prep_kernel<<<NPTS / 256, 256, 0, stream>>>(x, sq, apb, anb);

    dim3 grid(COLSPLIT, NPTS / ROWS_PER_BLOCK);  // (4, 128) = 512 blocks
    tile_kernel<<<grid, 32 * WAVES, 0, stream>>>(x, tgt, sq, apb, anb);

    final_kernel<<<1, 256, 0, stream>>>(apb, anb, out);
}